// MultiHeadSeqAttention_16363825398643
// MI455X (gfx1250) — compile-verified
//
#include <hip/hip_runtime.h>

typedef __attribute__((ext_vector_type(2))) float v2f;
typedef __attribute__((ext_vector_type(8))) float v8f;
typedef __attribute__((ext_vector_type(4))) int   v4i;

#define WMMA_F32(a, b, c) \
    __builtin_amdgcn_wmma_f32_16x16x4_f32(false, (a), false, (b), (short)0, (c), false, false)

// Problem constants (match reference)
constexpr int   Bb    = 8;
constexpr int   Mm    = 512;
constexpr int   Ll    = 1024;
constexpr int   Hh    = 512;
constexpr int   Kk    = 8;
constexpr int   Dd    = 64;
constexpr int   SKV   = Mm + Ll;   // 1536
constexpr int   BH    = Bb * Kk;   // 64 batch-heads
constexpr int   NB    = 68;        // key blocks per 64-query super-tile (covers m+L-1)
constexpr float SCALE = 0.125f;    // 1/sqrt(64)
constexpr float RAMPF = 32.0f;

// ---------------------------------------------------------------------------
// Async global->LDS staging (CDNA5 GLOBAL_LOAD_ASYNC_TO_LDS_B128, ASYNCcnt).
// Builtin expects (int4 AS1*, int4 AS3*, imm offset, imm cpol) — offset must
// be an integer constant expression, hence the template parameter.
// Falls back to a synchronous copy if the builtin is unavailable.
// ---------------------------------------------------------------------------
#if defined(__has_builtin)
#if __has_builtin(__builtin_amdgcn_global_load_async_to_lds_b128)
#define USE_ASYNC_LDS 1
#endif
#endif

typedef __attribute__((address_space(1))) v4i gv4i;
typedef __attribute__((address_space(3))) v4i lv4i;

template <int OFF>
__device__ __forceinline__ void async_copy16(const float* __restrict__ g, float* l)
{
#ifdef USE_ASYNC_LDS
    __builtin_amdgcn_global_load_async_to_lds_b128((gv4i*)g, (lv4i*)l, OFF, 0);
#else
    *(float4*)((char*)l + OFF) = *(const float4*)((const char*)g + OFF);
#endif
}

// ---------------------------------------------------------------------------
// Projection GEMM: Y = X @ W^T  (X:[rows,H], W:[H,H] row-major [out,in])
// HS=1 -> write to head layout [(b*K+h), s, d]; else flat [row, H].
// One wave computes a 32x32 output tile (2x2 register blocking): each k-step
// loads 2 A + 2 B fragments (32 B/lane) and issues 4 WMMAs -> 2x the
// register-level arithmetic intensity of a 16x16 tile.
// ---------------------------------------------------------------------------
template <int S, int HS>
__global__ void proj_gemm(const float* __restrict__ X, const float* __restrict__ W,
                          float* __restrict__ Y, int rows)
{
    const int lane = threadIdx.x & 31;
    const int wid  = threadIdx.x >> 5;
    const int tile = blockIdx.x * 4 + wid;
    const int ctiles = Hh / 32;                 // 16
    const int ntiles = (rows >> 5) * ctiles;
    if (tile >= ntiles) return;                 // wave-uniform guard (EXEC stays full)
    const int rt = tile / ctiles;
    const int ct = tile % ctiles;
    const int lo = lane & 15, hi = lane >> 4;

    const float* xrow0 = X + (size_t)(rt * 32 + lo) * Hh;        // A rows m    = lane%16
    const float* xrow1 = X + (size_t)(rt * 32 + 16 + lo) * Hh;   // A rows m+16
    const float* wrow0 = W + (size_t)(ct * 32 + lo) * Hh;        // B cols n    (W^T[k][n]=W[n][k])
    const float* wrow1 = W + (size_t)(ct * 32 + 16 + lo) * Hh;   // B cols n+16

    v8f c00 = {0.f,0.f,0.f,0.f,0.f,0.f,0.f,0.f};
    v8f c01 = c00, c10 = c00, c11 = c00;
#pragma unroll 8
    for (int k0 = 0; k0 < Hh; k0 += 4) {
        v2f a0 = *(const v2f*)(xrow0 + k0 + 2 * hi);   // k = k0 + vgpr + 2*hi
        v2f a1 = *(const v2f*)(xrow1 + k0 + 2 * hi);
        v2f b0 = *(const v2f*)(wrow0 + k0 + 2 * hi);
        v2f b1 = *(const v2f*)(wrow1 + k0 + 2 * hi);
        c00 = WMMA_F32(a0, b0, c00);
        c01 = WMMA_F32(a0, b1, c01);
        c10 = WMMA_F32(a1, b0, c10);
        c11 = WMMA_F32(a1, b1, c11);
    }

#pragma unroll
    for (int v = 0; v < 8; ++v) {
        const int g0   = rt * 32 + v + 8 * hi;       // rows of c0x
        const int g1   = g0 + 16;                    // rows of c1x
        const int col0 = ct * 32 + lo;               // cols of cx0
        const int col1 = col0 + 16;                  // cols of cx1
        if (HS) {
            const int b0i = g0 / S, s0 = g0 % S;     // S compile-time: no runtime div
            const int b1i = g1 / S, s1 = g1 % S;
            Y[(((size_t)(b0i * Kk + (col0 >> 6))) * S + s0) * Dd + (col0 & 63)] = c00[v];
            Y[(((size_t)(b0i * Kk + (col1 >> 6))) * S + s0) * Dd + (col1 & 63)] = c01[v];
            Y[(((size_t)(b1i * Kk + (col0 >> 6))) * S + s1) * Dd + (col0 & 63)] = c10[v];
            Y[(((size_t)(b1i * Kk + (col1 >> 6))) * S + s1) * Dd + (col1 & 63)] = c11[v];
        } else {
            Y[(size_t)g0 * Hh + col0] = c00[v];
            Y[(size_t)g0 * Hh + col1] = c01[v];
            Y[(size_t)g1 * Hh + col0] = c10[v];
            Y[(size_t)g1 * Hh + col1] = c11[v];
        }
    }
}

// ---------------------------------------------------------------------------
// Positional scores: POS[bh, m, l] = sum_d Q[bh,m,d] * PE[d,l]
// (bound by the 128 MB store stream; 16x16 tiles suffice)
// ---------------------------------------------------------------------------
__global__ void pos_gemm(const float* __restrict__ Q, const float* __restrict__ PE,
                         float* __restrict__ POS)
{
    const int lane = threadIdx.x & 31;
    const int wid  = threadIdx.x >> 5;
    const int tile = blockIdx.x * 4 + wid;       // 64 * 32 * 64 tiles
    const int lt = tile & 63;
    const int mt = (tile >> 6) & 31;
    const int bh = tile >> 11;
    const int lo = lane & 15, hi = lane >> 4;

    const float* qrow  = Q + ((size_t)bh * Mm + mt * 16 + lo) * Dd;
    const float* pecol = PE + lt * 16 + lo;

    v8f c = {0.f, 0.f, 0.f, 0.f, 0.f, 0.f, 0.f, 0.f};
#pragma unroll
    for (int k0 = 0; k0 < Dd; k0 += 4) {
        v2f a = *(const v2f*)(qrow + k0 + 2 * hi);
        v2f b;
        b.x = pecol[(size_t)(k0 + 2 * hi) * Ll];
        b.y = pecol[(size_t)(k0 + 2 * hi + 1) * Ll];
        c = WMMA_F32(a, b, c);
    }
#pragma unroll
    for (int v = 0; v < 8; ++v) {
        const int m = mt * 16 + v + 8 * hi;
        POS[((size_t)bh * Mm + m) * Ll + lt * 16 + lo] = c[v];
    }
}

// ---------------------------------------------------------------------------
// Fused sliding-window attention, flash-style streaming softmax.
// Block = 4 waves = 4 consecutive 16-query tiles of one batch-head (64 rows).
// All waves share one absolute key range [mBase, mBase+NB*16); K/V tiles are
// double-buffered in LDS via async global->LDS copies (4x traffic reduction
// vs per-wave loads); band masking handles the per-row window.
// ---------------------------------------------------------------------------
__global__ void attn_fused(const float* __restrict__ Q, const float* __restrict__ Kh,
                           const float* __restrict__ Vh, const float* __restrict__ POS,
                           const float* __restrict__ span_val, float* __restrict__ O)
{
    __shared__ __align__(16) float kbuf[2][16 * 64];
    __shared__ __align__(16) float vbuf[2][16 * 64];
    __shared__ float pbuf[4][16 * 16];

    const int tid  = threadIdx.x;
    const int lane = tid & 31;
    const int wid  = tid >> 5;
    const int lo = lane & 15, hi = lane >> 4;

    const int bh    = blockIdx.x >> 3;     // 64 batch-heads
    const int mt4   = blockIdx.x & 7;      // 8 super-tiles of 64 queries
    const int mBase = mt4 * 64;
    const int m0    = mBase + wid * 16;    // this wave's query tile
    const int b = bh >> 3, h = bh & 7;
    const float span = span_val[h];
    float* P = pbuf[wid];

    const float* kbase  = Kh  + ((size_t)bh * SKV + mBase) * Dd;
    const float* vbase  = Vh  + ((size_t)bh * SKV + mBase) * Dd;
    const float* posrow = POS + (size_t)bh * Mm * Ll;

    // Preload full Q tile as A fragments (16 rows x 64 k -> 16 chunks of K=4)
    v2f qa[16];
    const float* qrow = Q + ((size_t)bh * Mm + m0 + lo) * Dd;
#pragma unroll
    for (int dc = 0; dc < 16; ++dc)
        qa[dc] = *(const v2f*)(qrow + dc * 4 + 2 * hi);

    v8f acc0 = {0.f,0.f,0.f,0.f,0.f,0.f,0.f,0.f};
    v8f acc1 = acc0, acc2 = acc0, acc3 = acc0;
    float rowmax[8], rowsum[8];
#pragma unroll
    for (int v = 0; v < 8; ++v) { rowmax[v] = -__builtin_inff(); rowsum[v] = 0.f; }

    // Stage block 0: 128 threads copy 16x64 K and V tiles (32 B per thread each)
    {
        const float* gK = kbase + tid * 8;
        const float* gV = vbase + tid * 8;
        async_copy16<0> (gK, &kbuf[0][tid * 8]);
        async_copy16<16>(gK, &kbuf[0][tid * 8]);
        async_copy16<0> (gV, &vbuf[0][tid * 8]);
        async_copy16<16>(gV, &vbuf[0][tid * 8]);
    }

    for (int blk = 0; blk < NB; ++blk) {
        const int cur = blk & 1;

        if (blk + 1 < NB) {                       // prefetch next block
            const size_t off = (size_t)(blk + 1) * 16 * Dd;
            const float* gK = kbase + off + tid * 8;
            const float* gV = vbase + off + tid * 8;
            async_copy16<0> (gK, &kbuf[cur ^ 1][tid * 8]);
            async_copy16<16>(gK, &kbuf[cur ^ 1][tid * 8]);
            async_copy16<0> (gV, &vbuf[cur ^ 1][tid * 8]);
            async_copy16<16>(gV, &vbuf[cur ^ 1][tid * 8]);
#ifdef USE_ASYNC_LDS
            asm volatile("s_wait_asynccnt 0x4" ::: "memory");   // current block landed
#endif
        } else {
#ifdef USE_ASYNC_LDS
            asm volatile("s_wait_asynccnt 0x0" ::: "memory");
#endif
        }
        __syncthreads();                          // staged tiles visible to all waves

        const float* kc = kbuf[cur];
        const float* vc = vbuf[cur];

        // ---- S = Q @ K^T (16x16 over K=64) from LDS ----
        v8f s = {0.f,0.f,0.f,0.f,0.f,0.f,0.f,0.f};
#pragma unroll
        for (int dc = 0; dc < 16; ++dc) {
            v2f kb = *(const v2f*)(kc + lo * 64 + dc * 4 + 2 * hi);
            s = WMMA_F32(qa[dc], kb, s);
        }

        // ---- add POS, scale, band-mask, online softmax ----
        float corr[8];
#pragma unroll
        for (int v = 0; v < 8; ++v) {
            const int r = v + 8 * hi;
            const int m = m0 + r;
            const int l = (blk - wid) * 16 + lo - r;       // relative position n-m
            const int lc = l < 0 ? 0 : (l > Ll - 1 ? Ll - 1 : l);
            const float pv = posrow[(size_t)m * Ll + lc];
            const bool valid = (l >= 0) & (l < Ll);
            const float sc = valid ? (s[v] + pv) * SCALE : -__builtin_inff();

            float bm = sc;                              // row max within block
            bm = fmaxf(bm, __shfl_xor(bm, 1, 32));
            bm = fmaxf(bm, __shfl_xor(bm, 2, 32));
            bm = fmaxf(bm, __shfl_xor(bm, 4, 32));
            bm = fmaxf(bm, __shfl_xor(bm, 8, 32));
            const float nm = fmaxf(rowmax[v], bm);
            // guard: fully-masked block while row untouched (nm=-inf) -> no NaN
            const float cr = (nm == -__builtin_inff()) ? 1.0f : __expf(rowmax[v] - nm);
            const float p  = valid ? __expf(sc - nm) : 0.0f;
            float ps = p;                               // row sum within block
            ps += __shfl_xor(ps, 1, 32);
            ps += __shfl_xor(ps, 2, 32);
            ps += __shfl_xor(ps, 4, 32);
            ps += __shfl_xor(ps, 8, 32);
            rowsum[v] = rowsum[v] * cr + ps;
            rowmax[v] = nm;
            corr[v]   = cr;

            // post-softmax span mask (numerator only)
            float mk = ((float)(l - (Ll - 1)) + span * (float)Ll) * (1.0f / RAMPF) + 1.0f;
            mk = mk < 0.f ? 0.f : (mk > 1.f ? 1.f : mk);
            P[r * 16 + lo] = p * mk;
        }
#pragma unroll
        for (int v = 0; v < 8; ++v) {
            acc0[v] *= corr[v]; acc1[v] *= corr[v];
            acc2[v] *= corr[v]; acc3[v] *= corr[v];
        }

        // C-layout P tile -> A fragments via per-wave LDS slice
        asm volatile("s_wait_dscnt 0x0" ::: "memory");
        v2f pa[4];
#pragma unroll
        for (int kc2 = 0; kc2 < 4; ++kc2)
            pa[kc2] = *(const v2f*)(P + lo * 16 + kc2 * 4 + 2 * hi);

        // ---- acc += P @ V (16x64 over K=16) from LDS ----
#pragma unroll
        for (int kc2 = 0; kc2 < 4; ++kc2) {
            const int r0 = kc2 * 4 + 2 * hi;            // B: k row = vgpr + 2*hi
            v2f vb0, vb1, vb2, vb3;
            vb0.x = vc[r0 * 64 +  0 + lo]; vb0.y = vc[(r0 + 1) * 64 +  0 + lo];
            vb1.x = vc[r0 * 64 + 16 + lo]; vb1.y = vc[(r0 + 1) * 64 + 16 + lo];
            vb2.x = vc[r0 * 64 + 32 + lo]; vb2.y = vc[(r0 + 1) * 64 + 32 + lo];
            vb3.x = vc[r0 * 64 + 48 + lo]; vb3.y = vc[(r0 + 1) * 64 + 48 + lo];
            acc0 = WMMA_F32(pa[kc2], vb0, acc0);
            acc1 = WMMA_F32(pa[kc2], vb1, acc1);
            acc2 = WMMA_F32(pa[kc2], vb2, acc2);
            acc3 = WMMA_F32(pa[kc2], vb3, acc3);
        }
        __syncthreads();                          // done reading before next overwrite
    }

    // Epilogue: divide by softmax denominator, write [B, M, H] flat layout
#pragma unroll
    for (int v = 0; v < 8; ++v) {
        const int r = v + 8 * hi;
        const int m = m0 + r;
        const float inv = 1.0f / rowsum[v];
        const size_t base = ((size_t)b * Mm + m) * Hh + h * Dd;
        O[base +  0 + lo] = acc0[v] * inv;
        O[base + 16 + lo] = acc1[v] * inv;
        O[base + 32 + lo] = acc2[v] * inv;
        O[base + 48 + lo] = acc3[v] * inv;
    }
}

// ---------------------------------------------------------------------------
extern "C" void kernel_launch(void* const* d_in, const int* in_sizes, int n_in,
                              void* d_out, int out_size, void* d_ws, size_t ws_size,
                              hipStream_t stream)
{
    const float* query  = (const float*)d_in[0];
    const float* key    = (const float*)d_in[1];
    const float* value  = (const float*)d_in[2];
    const float* key_pe = (const float*)d_in[3];
    const float* Wq     = (const float*)d_in[4];
    const float* Wk     = (const float*)d_in[5];
    const float* Wv     = (const float*)d_in[6];
    const float* Wo     = (const float*)d_in[7];
    const float* span   = (const float*)d_in[8];

    float* ws    = (float*)d_ws;
    float* qh    = ws;                                  //  8 MB  [BH, M,   D]
    float* kh    = qh   + (size_t)BH * Mm  * Dd;        // 24 MB  [BH, M+L, D]
    float* vh    = kh   + (size_t)BH * SKV * Dd;        // 24 MB  [BH, M+L, D]
    float* pos   = vh   + (size_t)BH * SKV * Dd;        // 128 MB [BH, M, L]
    float* aflat = pos  + (size_t)BH * Mm  * Ll;        //  8 MB  [B, M, H]

    const dim3 blk(128);   // 4 waves per block

    // QKV projections (head-split layouts), 32x32 tiles per wave
    proj_gemm<Mm, 1><<<((Bb * Mm  / 32) * (Hh / 32)) / 4, blk, 0, stream>>>(query, Wq, qh, Bb * Mm);
    proj_gemm<SKV, 1><<<((Bb * SKV / 32) * (Hh / 32)) / 4, blk, 0, stream>>>(key,   Wk, kh, Bb * SKV);
    proj_gemm<SKV, 1><<<((Bb * SKV / 32) * (Hh / 32)) / 4, blk, 0, stream>>>(value, Wv, vh, Bb * SKV);

    // Relative positional scores (shared pe, per batch-head q)
    pos_gemm<<<(BH * (Mm / 16) * (Ll / 16)) / 4, blk, 0, stream>>>(qh, key_pe, pos);

    // Fused streaming attention -> [B, M, H]; one block per 64-query super-tile
    attn_fused<<<BH * (Mm / 64), blk, 0, stream>>>(qh, kh, vh, pos, span, aflat);

    // Output projection straight into d_out
    proj_gemm<Mm, 0><<<((Bb * Mm / 32) * (Hh / 32)) / 4, blk, 0, stream>>>(aflat, Wo, (float*)d_out, Bb * Mm);
}